// KnowledgeAwareGraphNetwork_2637109919866
// MI455X (gfx1250) — compile-verified
//
#include <hip/hip_runtime.h>

#define NN 50000
#define NE 800000
#define DIN 128
#define DHID 256
#define DOUT 128

typedef __attribute__((ext_vector_type(2))) float v2f;
typedef __attribute__((ext_vector_type(8))) float v8f;

__global__ void zero_kernel(float* __restrict__ p, int n4) {
  int i = blockIdx.x * blockDim.x + threadIdx.x;
  if (i < n4) ((float4*)p)[i] = make_float4(0.f, 0.f, 0.f, 0.f);
}

// feat[node, :] = emb[cncpt_ids[node], :], one float4 per thread (32 per node)
__global__ void gather_kernel(const int* __restrict__ ids,
                              const float* __restrict__ emb,
                              float* __restrict__ feat) {
  int t = blockIdx.x * blockDim.x + threadIdx.x;
  int node = t >> 5;
  int c = t & 31;
  if (node < NN) {
    long long cid = (long long)ids[node];
    ((float4*)feat)[(long long)node * 32 + c] =
        ((const float4*)emb)[cid * 32 + c];
  }
}

// agg[dst[e], :] += h[src[e], :]   (D = 4 << D4SHIFT floats per row)
// consecutive threads = consecutive float4s of one edge -> coalesced row read,
// atomics land on consecutive addresses (L2-resident, one cacheline per 16 lanes)
template <int D4SHIFT>
__global__ void scatter_add_kernel(const float* __restrict__ h,
                                   const int* __restrict__ src,
                                   const int* __restrict__ dst,
                                   float* __restrict__ agg) {
  long long t = (long long)blockIdx.x * blockDim.x + threadIdx.x;
  int e = (int)(t >> D4SHIFT);
  int c = (int)(t & ((1 << D4SHIFT) - 1));
  if (e < NE) {
    const int D = 4 << D4SHIFT;
    long long s = (long long)src[e];
    long long d = (long long)dst[e];
    float4 v = ((const float4*)(h + s * D))[c];
    float* o = agg + d * D + (c << 2);
    atomicAdd(o + 0, v.x);
    atomicAdd(o + 1, v.y);
    atomicAdd(o + 2, v.z);
    atomicAdd(o + 3, v.w);
  }
}

// out[M,N] = relu(A[M,K] @ W[K,N] + bias[N]) via V_WMMA_F32_16X16X4_F32.
// Block = 4 waves, each wave owns one 16x16 C tile (4 consecutive M tiles,
// shared N tile). W[:, ntile*16 .. +16) staged in LDS once per block.
template <int K, int N>
__global__ __launch_bounds__(128) void gemm_bias_relu(
    const float* __restrict__ A, const float* __restrict__ W,
    const float* __restrict__ bias, float* __restrict__ out, int Mtiles) {
  __shared__ float lw[K * 16];
  const int nbase = blockIdx.x * 16;

  for (int i = threadIdx.x; i < K * 16; i += 128) {
    int k = i >> 4;
    int n = i & 15;
    lw[i] = W[k * N + nbase + n];
  }
  __syncthreads();

  const int wave = threadIdx.x >> 5;
  const int lane = threadIdx.x & 31;
  const int mtile = blockIdx.y * 4 + wave;
  if (mtile >= Mtiles) return;  // whole-wave uniform exit; EXEC stays all-1s

  const int mbase = mtile * 16;
  const int n16 = lane & 15;
  const int ks = (lane >> 4) * 2;  // A/B fragment K sub-offset: 0 or 2
  const float* arow = A + (long long)(mbase + n16) * K + ks;

  v8f acc = {};
  for (int kk = 0; kk < K; kk += 4) {
    // A 16x4 frag: lanes 0-15 -> K = kk+{0,1}; lanes 16-31 -> K = kk+{2,3}
    v2f a = *(const v2f*)(arow + kk);
    // B 4x16 frag from LDS: v0 = row kk+ks, v1 = row kk+ks+1, N = lane&15
    v2f b;
    b.x = lw[(kk + ks) * 16 + n16];
    b.y = lw[(kk + ks + 1) * 16 + n16];
    acc = __builtin_amdgcn_wmma_f32_16x16x4_f32(
        false, a, false, b, (short)0, acc, false, false);
  }

  // C/D layout: VGPR v -> M = mbase + v + (lane>=16 ? 8 : 0), N = nbase + (lane&15)
  const float bv = bias[nbase + n16];
  const int moff = (lane >> 4) * 8;
#pragma unroll
  for (int v = 0; v < 8; ++v) {
    float x = acc[v] + bv;
    out[(long long)(mbase + moff + v) * N + nbase + n16] = fmaxf(x, 0.f);
  }
}

extern "C" void kernel_launch(void* const* d_in, const int* in_sizes, int n_in,
                              void* d_out, int out_size, void* d_ws, size_t ws_size,
                              hipStream_t stream) {
  const int*   ids = (const int*)d_in[0];
  const int*   src = (const int*)d_in[1];
  const int*   dst = (const int*)d_in[2];
  const float* emb = (const float*)d_in[3];
  const float* W1  = (const float*)d_in[4];
  const float* b1  = (const float*)d_in[5];
  const float* W2  = (const float*)d_in[6];
  const float* b2  = (const float*)d_in[7];
  float* out = (float*)d_out;

  // ws layout (floats): region A = NN*256 (feat NN*128 | agg1 NN*128, later agg2 NN*256)
  //                     region B = NN*256 (h1)
  float* regA = (float*)d_ws;
  float* feat = regA;
  float* agg1 = regA + (size_t)NN * DIN;
  float* agg2 = regA;  // recycled after gemm1 consumes agg1
  float* h1   = regA + (size_t)NN * 2 * DIN;

  const int Mtiles = NN / 16;  // 3125, exact

  // Layer 1
  {
    int n4 = NN * DIN / 4;
    zero_kernel<<<(n4 + 255) / 256, 256, 0, stream>>>(agg1, n4);
  }
  {
    long long t = (long long)NN * (DIN / 4);
    gather_kernel<<<(unsigned)((t + 255) / 256), 256, 0, stream>>>(ids, emb, feat);
  }
  {
    long long t = (long long)NE * (DIN / 4);  // 25.6M threads, 1 wave per edge
    scatter_add_kernel<5><<<(unsigned)((t + 255) / 256), 256, 0, stream>>>(feat, src, dst, agg1);
  }
  {
    dim3 grid(DHID / 16, (Mtiles + 3) / 4);
    gemm_bias_relu<DIN, DHID><<<grid, 128, 0, stream>>>(agg1, W1, b1, h1, Mtiles);
  }

  // Layer 2 (agg2 overlaps feat/agg1 — both dead now; stream order serializes)
  {
    int n4 = NN * DHID / 4;
    zero_kernel<<<(n4 + 255) / 256, 256, 0, stream>>>(agg2, n4);
  }
  {
    long long t = (long long)NE * (DHID / 4);
    scatter_add_kernel<6><<<(unsigned)((t + 255) / 256), 256, 0, stream>>>(h1, src, dst, agg2);
  }
  {
    dim3 grid(DOUT / 16, (Mtiles + 3) / 4);
    gemm_bias_relu<DHID, DOUT><<<grid, 128, 0, stream>>>(agg2, W2, b2, out, Mtiles);
  }
}